// X_50259707298065
// MI455X (gfx1250) — compile-verified
//
#include <hip/hip_runtime.h>
#include <stdint.h>

// out[c, n, b] = x_c[n ^ 2048, b]  for c in {real, imag}
// Reference operator = X (x) I^11 on a 2^12 state: flip bit 11 of the row
// index -> pure permutation, 0 FLOPs. Traffic = 16 MB in + 16 MB out,
// HBM-roofline ~1.4 us at 23.3 TB/s.
//
// CDNA5 async data-mover implementation (ASYNCcnt-tracked LDS bounce):
//  - per block: one scalar real/imag select (component is block-uniform),
//  - row-flip folded into a single XOR of bit 18 of the float4 index,
//  - 4 async b128 loads sharing ONE address VGPR pair, slabs selected by the
//    instruction offset: immediate (which advances global AND LDS address),
//  - one s_wait_asynccnt, then 4 async stores, dst linear.

#define BLK     256
#define UNROLL  4
// 2 components * 4096 rows * 128 float4/row = 1048576 float4 transfers
#define NBLOCKS 1024            // 1048576 / (BLK*UNROLL)

__global__ __launch_bounds__(BLK) void qshift_permute_async(
    const float* __restrict__ xr,
    const float* __restrict__ xi,
    float* __restrict__ out)
{
    __shared__ float4 smem[BLK * UNROLL];   // 16 KB bounce buffer per block

    const unsigned tid = threadIdx.x;
    const unsigned blk = blockIdx.x;

    // Component is uniform per block: blocks [0,512) -> real, [512,1024) -> imag.
    const float* __restrict__ srcComp = (blk < 512u) ? xr : xi;

    // float4 index within the component; row-flip == XOR bit 18.
    const unsigned ubase = ((blk & 511u) << 10) + tid;
    const float* src = srcComp + ((size_t)(ubase ^ 0x40000u) << 2);  // *4 floats

    // Output is linear in the global float4 index.
    float* dst = out + ((((size_t)blk << 10) + tid) << 2);

    // Low 32 bits of a generic LDS pointer == LDS byte offset (compiler folds
    // this to tid*16).
    const unsigned lds = (unsigned)(uintptr_t)&smem[tid];

    // The offset: immediate is added to BOTH the global and the LDS address,
    // and our LDS slab stride is exactly BLK*16 = 4096 bytes. Input is
    // streamed once -> non-temporal loads keep L2 clean.
#define ALOAD(OFF)                                                            \
    asm volatile("global_load_async_to_lds_b128 %0, %1, off offset:" #OFF     \
                 " th:TH_LOAD_NT"                                             \
                 :: "v"(lds), "v"(src) : "memory")
#define ASTORE(OFF)                                                           \
    asm volatile("global_store_async_from_lds_b128 %0, %1, off offset:" #OFF  \
                 :: "v"(dst), "v"(lds) : "memory")

    ALOAD(0);
    ALOAD(4096);
    ALOAD(8192);
    ALOAD(12288);

    // All 4 loads complete (ASYNCcnt in order) before stores read LDS.
    asm volatile("s_wait_asynccnt 0" ::: "memory");

    ASTORE(0);
    ASTORE(4096);
    ASTORE(8192);
    ASTORE(12288);
    // s_endpgm's implicit wait-idle drains the stores before LDS dealloc.

#undef ALOAD
#undef ASTORE
}

extern "C" void kernel_launch(void* const* d_in, const int* in_sizes, int n_in,
                              void* d_out, int out_size, void* d_ws, size_t ws_size,
                              hipStream_t stream)
{
    (void)in_sizes; (void)n_in; (void)out_size; (void)d_ws; (void)ws_size;

    const float* xr = (const float*)d_in[0];
    const float* xi = (const float*)d_in[1];
    float* out      = (float*)d_out;

    dim3 grid(NBLOCKS);   // 1024 blocks
    dim3 block(BLK);      // 256 threads = 8 wave32
    hipLaunchKernelGGL(qshift_permute_async, grid, block, 0, stream, xr, xi, out);
}